// HiPPOScale_47287589929890
// MI455X (gfx1250) — compile-verified
//
#include <hip/hip_runtime.h>
#include <hip/hip_bf16.h>

// HiPPO-LegS bilinear scan on MI455X (gfx1250, wave32).
//   c_t = c_{t-1} @ A_t^T + f_t[:,None] * B_t     (batch=512, N=256, L=1024)
// Batch rows are independent -> each block owns 16 batch rows and loops over t.
// Per step, a 512x256 x 256x256 f32 matmul done with V_WMMA_F32_16X16X4_F32.

typedef __attribute__((ext_vector_type(2))) float v2f;
typedef __attribute__((ext_vector_type(8))) float v8f;

#define L_STEPS   1024
#define BATCH     512
#define NDIM      256
#define BTILE     16          // batch rows per block
#define WAVES     16          // 16 waves * 16 cols = 256 output columns
#define ROWSTRIDE 260         // 256 + 4 floats pad: LDS banks hit = 4*lane + k -> conflict-free b64

__global__ __launch_bounds__(WAVES * 32)
void hippo_scan_wmma(const float* __restrict__ inputs,   // (L, BATCH)
                     const float* __restrict__ A,        // (L, N, N) row-major
                     const float* __restrict__ Bvec,     // (L, N)
                     float* __restrict__ out)            // (L, BATCH, N)
{
    // Double-buffered recurrent state: c tile (16 batch rows x 256), 33 KB total.
    __shared__ float cbuf[2][BTILE * ROWSTRIDE];

    const int tid  = threadIdx.x;
    const int lane = tid & 31;
    const int wave = tid >> 5;          // 0..15
    const int n0   = wave << 4;         // this wave's output-column tile
    const int m0   = blockIdx.x << 4;   // this block's batch-row tile
    const int l15  = lane & 15;
    const int hi   = lane >> 4;         // 0: lanes 0-15 (K=0,1), 1: lanes 16-31 (K=2,3)
    const int koff = hi * 2;

    // c0 = 0
    for (int i = tid; i < 2 * BTILE * ROWSTRIDE; i += blockDim.x) {
        cbuf[0][i] = 0.0f;              // flattened: zeroes both buffers
    }
    __syncthreads();

    // Per-lane invariant addressing.
    // WMMA "B" operand tile (A_t^T, 4x16): element B[k][n] = A_t[n][k].
    //   lane l<16 supplies K=k0,k0+1 for column n0+l; lanes>=16 supply K=k0+2,k0+3.
    //   -> one b64 global load per lane per K-tile, from row (n0+l15), col k0+koff.
    const size_t aRowOff = (size_t)(n0 + l15) * NDIM + (size_t)koff;
    // WMMA "A" operand tile (c_old, 16x4): lane l<16 supplies row M=l, K=k0,k0+1;
    //   lanes>=16 row M=l-16, K=k0+2,k0+3 -> one b64 LDS load per lane per K-tile.
    const int cRowOff = l15 * ROWSTRIDE + koff;

    int cur = 0;
    for (int t = 0; t < L_STEPS; ++t) {
        const float* __restrict__ Arow = A + ((size_t)t << 16) + aRowOff;
        const float* __restrict__ cRow = &cbuf[cur][cRowOff];

        // Prefetch next step's A_t slice while we compute this step.
        if (t + 1 < L_STEPS) {
            __builtin_prefetch(Arow + (1 << 16), 0, 0);   // global_prefetch_b8
        }

        // Seed accumulator with the rank-1 term: acc[b][n] = f_t[b] * B_t[n].
        // C/D layout: VGPR i, lanes 0-15 -> M=i, N=lane; lanes 16-31 -> M=i+8.
        const float bt = Bvec[(size_t)t * NDIM + n0 + l15];
        const float* __restrict__ frow = inputs + (size_t)t * BATCH + m0 + hi * 8;
        v8f acc;
#pragma unroll
        for (int i = 0; i < 8; ++i) {
            acc[i] = frow[i] * bt;
        }

        // acc += c_old(16x256) @ A_t^T(256x16), K-tiled by 4.
#pragma unroll 4
        for (int k = 0; k < NDIM; k += 4) {
            v2f a = *(const v2f*)(cRow + k);   // ds_load_b64 (c_old tile)
            v2f b = *(const v2f*)(Arow + k);   // global_load_b64 (A_t^T tile)
            acc = __builtin_amdgcn_wmma_f32_16x16x4_f32(
                /*neg_a=*/false, a, /*neg_b=*/false, b,
                /*c_mod=*/(short)0, acc, /*reuse_a=*/false, /*reuse_b=*/false);
        }

        // Write c_new into the other LDS buffer and stream it to global output.
        const int nxt = cur ^ 1;
        const int nb  = n0 + l15;
        float* __restrict__ outT = out + ((size_t)t * BATCH + m0) * NDIM;
#pragma unroll
        for (int i = 0; i < 8; ++i) {
            const int bl = i + hi * 8;                 // local batch row 0..15
            cbuf[nxt][bl * ROWSTRIDE + nb] = acc[i];   // ds_store_b32
            outT[(size_t)bl * NDIM + nb]  = acc[i];    // coalesced global_store_b32
        }

        __syncthreads();   // covers RAW on nxt buffer and WAR on cur buffer
        cur = nxt;
    }
}

extern "C" void kernel_launch(void* const* d_in, const int* in_sizes, int n_in,
                              void* d_out, int out_size, void* d_ws, size_t ws_size,
                              hipStream_t stream) {
    const float* inputs = (const float*)d_in[0];   // (1024, 512)
    const float* A      = (const float*)d_in[1];   // (1024, 256, 256)
    const float* Bvec   = (const float*)d_in[2];   // (1024, 256)
    float* out          = (float*)d_out;           // (1024, 512, 256)

    dim3 grid(BATCH / BTILE);       // 32 blocks, one per 16-row batch tile
    dim3 block(WAVES * 32);         // 512 threads = 16 wave32 waves
    hippo_scan_wmma<<<grid, block, 0, stream>>>(inputs, A, Bvec, out);
}